// RHGNNLayer_77129022701794
// MI455X (gfx1250) — compile-verified
//
#include <hip/hip_runtime.h>
#include <hip/hip_bf16.h>

typedef __attribute__((ext_vector_type(2))) float v2f;
typedef __attribute__((ext_vector_type(8))) float v8f;

#define NNODES 50000
#define NEDGES 200000
#define INDIM 256
#define HEADS 8
#define HID 32
#define HD 256            // HEADS*HID
#define NEG_SLOPE 0.2f
#define AS_STRIDE 34      // padded LDS row stride (floats): conflict-free, 8B aligned

// ---------------------------------------------------------------------------
// Shared GEMM core: one block (128 threads = 4 waves) computes the 16-row
// strip [m0, m0+16) x all 256 output columns of C = A(Mx256) * B(256x256),
// full fp32 via V_WMMA_F32_16X16X4_F32.
//   wave w owns columns [64w, 64w+64) as 4 accumulators (16x16 tiles).
//   A 16x32 K-panels staged in LDS (coalesced float4 loads, padded stride).
//   WMMA fragment layout (ISA 7.12.2): A/B VGPR v, lanegroup g -> K = v+2g;
//   C VGPR i -> row m0+i+8g, col n0+lane%16.
// No divergence anywhere -> EXEC all-ones for every WMMA.
// ---------------------------------------------------------------------------
__device__ __forceinline__ void gemm16x256_core(
    const float* __restrict__ A, const float* __restrict__ B,
    int m0, float* __restrict__ As, v8f acc[4])
{
    const int t = threadIdx.x;
    const int wave = t >> 5;
    const int lane = t & 31;
    const int l = lane & 15, g = lane >> 4;
    const int n0w = wave * 64;

    for (int kb = 0; kb < INDIM; kb += 32) {
        __syncthreads();                       // previous panel fully consumed
        {   // stage A[m0..m0+15][kb..kb+31] -> LDS, one float4 per thread
            const int r = t >> 3, q = t & 7;   // 16 rows x 8 quads
            const float4 v = *(const float4*)(A + (size_t)(m0 + r) * INDIM + kb + 4 * q);
            float* dp = As + r * AS_STRIDE + 4 * q;
            dp[0] = v.x; dp[1] = v.y; dp[2] = v.z; dp[3] = v.w;
        }
        __syncthreads();
#pragma unroll
        for (int kk = 0; kk < 32; kk += 4) {
            const int ka = kk + 2 * g;
            v2f a;
            a.x = As[l * AS_STRIDE + ka];      // ds_load, conflict-free rows
            a.y = As[l * AS_STRIDE + ka + 1];
            const float* b0 = B + (size_t)(kb + ka) * 256 + n0w + l;
            const float* b1 = b0 + 256;
#pragma unroll
            for (int j = 0; j < 4; ++j) {      // 4 WMMAs per A fragment
                v2f b;
                b.x = b0[16 * j];
                b.y = b1[16 * j];
                acc[j] = __builtin_amdgcn_wmma_f32_16x16x4_f32(
                    false, a, false, b, (short)0, acc[j], false, false);
            }
        }
    }
}

// Plain GEMM: C = A @ B
__global__ void __launch_bounds__(128) gemm256_wmma(
    const float* __restrict__ A, const float* __restrict__ B, float* __restrict__ C)
{
    __shared__ float As[16 * AS_STRIDE];
    const int wave = threadIdx.x >> 5;
    const int lane = threadIdx.x & 31;
    const int l = lane & 15, g = lane >> 4;
    const int m0 = blockIdx.x * 16;
    const int n0w = wave * 64;
    v8f acc[4] = {};
    gemm16x256_core(A, B, m0, As, acc);
#pragma unroll
    for (int j = 0; j < 4; ++j)
#pragma unroll
        for (int i = 0; i < 8; ++i)
            C[(size_t)(m0 + i + 8 * g) * 256 + n0w + 16 * j + l] = acc[j][i];
}

// Fused residual GEMM + gated combine:
//   conv = relu(agg)*sigmoid(alpha) + (feat@res_W + res_b)*(1-sigmoid(alpha))
// agg read from / conv written to the same buffer (relation's slice of d_out).
__global__ void __launch_bounds__(128) combine_res_wmma(
    const float* __restrict__ A, const float* __restrict__ B,
    const float* __restrict__ bias, const float* __restrict__ res_alpha, int dIdx,
    float* __restrict__ conv)
{
    __shared__ float As[16 * AS_STRIDE];
    const int wave = threadIdx.x >> 5;
    const int lane = threadIdx.x & 31;
    const int l = lane & 15, g = lane >> 4;
    const int m0 = blockIdx.x * 16;
    const int n0w = wave * 64;
    v8f acc[4] = {};
    gemm16x256_core(A, B, m0, As, acc);
    const float al = 1.f / (1.f + expf(-res_alpha[dIdx]));
#pragma unroll
    for (int j = 0; j < 4; ++j) {
        const float bcol = bias[n0w + 16 * j + l];
#pragma unroll
        for (int i = 0; i < 8; ++i) {
            const size_t idx = (size_t)(m0 + i + 8 * g) * 256 + n0w + 16 * j + l;
            const float agg = conv[idx];
            conv[idx] = fmaxf(agg, 0.f) * al + (acc[j][i] + bcol) * (1.f - al);
        }
    }
}

// ra = rel_emb[r] (64) @ rel_W[r] (64 x 512) -> 512 values
__global__ void compute_ra(const float* __restrict__ rel_emb_r,
                           const float* __restrict__ rel_W_r, float* __restrict__ ra)
{
    const int o = blockIdx.x * blockDim.x + threadIdx.x;
    if (o >= 512) return;
    float s = 0.f;
#pragma unroll 8
    for (int i = 0; i < 64; ++i) s += rel_emb_r[i] * rel_W_r[i * 512 + o];
    ra[o] = s;
}

// e_dst[n,h] = sum_d fdst[n,h,d]*ra[h,d] ; e_src[n,h] = sum_d fsrc[n,h,d]*ra[h,D+d]
__global__ void compute_e(const float* __restrict__ fsrc, const float* __restrict__ fdst,
                          const float* __restrict__ ra,
                          float* __restrict__ e_src, float* __restrict__ e_dst)
{
    const int id = blockIdx.x * blockDim.x + threadIdx.x;
    if (id >= NNODES * HEADS) return;
    const int n = id >> 3, h = id & 7;
    const float* ps = fsrc + (size_t)n * HD + h * HID;
    const float* pd = fdst + (size_t)n * HD + h * HID;
    const float* rh = ra + h * (2 * HID);
    float ss = 0.f, ds = 0.f;
#pragma unroll 8
    for (int d = 0; d < HID; ++d) {
        ds += pd[d] * rh[d];
        ss += ps[d] * rh[HID + d];
    }
    e_src[id] = ss;
    e_dst[id] = ds;
}

__global__ void fill_f32(float* __restrict__ p, float v, int n)
{
    const int i = blockIdx.x * blockDim.x + threadIdx.x;
    if (i < n) p[i] = v;
}
__global__ void fill_u32(unsigned* __restrict__ p, unsigned v, int n)
{
    const int i = blockIdx.x * blockDim.x + threadIdx.x;
    if (i < n) p[i] = v;
}

// pass 1: edge logit + per-dst max (monotonic float->uint key, atomicMax)
__global__ void edge_logits(const int* __restrict__ src, const int* __restrict__ dst,
                            const float* __restrict__ e_src, const float* __restrict__ e_dst,
                            float* __restrict__ eedge, unsigned* __restrict__ nmax)
{
    const int id = blockIdx.x * blockDim.x + threadIdx.x;
    if (id >= NEDGES * HEADS) return;
    const int e = id >> 3, h = id & 7;
    const int s = src[e], d = dst[e];
    float v = e_src[s * HEADS + h] + e_dst[d * HEADS + h];
    v = v > 0.f ? v : NEG_SLOPE * v;
    eedge[id] = v;
    const unsigned bits = __float_as_uint(v);
    const unsigned key = (bits & 0x80000000u) ? ~bits : (bits | 0x80000000u);
    atomicMax(nmax + d * HEADS + h, key);
}

// pass 2: ex = exp(e - max), per-dst sum
__global__ void edge_expsum(const int* __restrict__ dst, float* __restrict__ eedge,
                            const unsigned* __restrict__ nmax, float* __restrict__ nsum)
{
    const int id = blockIdx.x * blockDim.x + threadIdx.x;
    if (id >= NEDGES * HEADS) return;
    const int e = id >> 3, h = id & 7;
    const int d = dst[e];
    const unsigned k = nmax[d * HEADS + h];
    const float m = __uint_as_float((k & 0x80000000u) ? (k & 0x7FFFFFFFu) : ~k);
    const float ex = expf(eedge[id] - m);
    eedge[id] = ex;
    atomicAdd(nsum + d * HEADS + h, ex);
}

// pass 3: scatter msg = fsrc[src] * a into agg via float atomics
__global__ void edge_scatter(const int* __restrict__ src, const int* __restrict__ dst,
                             const float* __restrict__ fsrc, const float* __restrict__ eedge,
                             const float* __restrict__ nsum, float* __restrict__ agg)
{
    const int id = blockIdx.x * blockDim.x + threadIdx.x;
    if (id >= NEDGES * HEADS) return;
    const int e = id >> 3, h = id & 7;
    const int s = src[e], d = dst[e];
    const float a = eedge[id] / nsum[d * HEADS + h];
    const float4* ps = (const float4*)(fsrc + (size_t)s * HD + h * HID);
    float* pa = agg + (size_t)d * HD + h * HID;
#pragma unroll
    for (int q = 0; q < 8; ++q) {
        const float4 v = ps[q];
        atomicAdd(pa + 4 * q + 0, v.x * a);
        atomicAdd(pa + 4 * q + 1, v.y * a);
        atomicAdd(pa + 4 * q + 2, v.z * a);
        atomicAdd(pa + 4 * q + 3, v.w * a);
    }
}

// Cross attention for dst ntype 1: group = (conv_out[0], conv_out[3]); produces
// node_out[0] and node_out[3] in place (each thread owns its own slots).
__global__ void cross_mix(float* __restrict__ c0, float* __restrict__ c3,
                          const float* __restrict__ cross_attn)
{
    const int id = blockIdx.x * blockDim.x + threadIdx.x;
    if (id >= NNODES * HEADS) return;
    const int n = id >> 3, h = id & 7;
    float* p0 = c0 + (size_t)n * HD + h * HID;
    float* p3 = c3 + (size_t)n * HD + h * HID;
    float x0[HID], x3[HID];
#pragma unroll
    for (int q = 0; q < 8; ++q) {
        const float4 v0 = ((const float4*)p0)[q];
        const float4 v3 = ((const float4*)p3)[q];
        x0[4 * q + 0] = v0.x; x0[4 * q + 1] = v0.y; x0[4 * q + 2] = v0.z; x0[4 * q + 3] = v0.w;
        x3[4 * q + 0] = v3.x; x3[4 * q + 1] = v3.y; x3[4 * q + 2] = v3.z; x3[4 * q + 3] = v3.w;
    }
    const float* ca0 = cross_attn + 0 * HD + h * HID;
    const float* ca3 = cross_attn + 3 * HD + h * HID;
    float s00 = 0.f, s01 = 0.f, s30 = 0.f, s31 = 0.f;
#pragma unroll 8
    for (int d = 0; d < HID; ++d) {
        s00 += x0[d] * ca0[d]; s01 += x3[d] * ca0[d];
        s30 += x0[d] * ca3[d]; s31 += x3[d] * ca3[d];
    }
    s00 = s00 > 0.f ? s00 : NEG_SLOPE * s00;
    s01 = s01 > 0.f ? s01 : NEG_SLOPE * s01;
    s30 = s30 > 0.f ? s30 : NEG_SLOPE * s30;
    s31 = s31 > 0.f ? s31 : NEG_SLOPE * s31;
    float m = fmaxf(s00, s01);
    float ea = expf(s00 - m), eb = expf(s01 - m), inv = 1.f / (ea + eb);
    const float w00 = ea * inv, w01 = eb * inv;
    m = fmaxf(s30, s31);
    ea = expf(s30 - m); eb = expf(s31 - m); inv = 1.f / (ea + eb);
    const float w30 = ea * inv, w31 = eb * inv;
#pragma unroll
    for (int q = 0; q < 8; ++q) {
        float4 o0, o3;
        o0.x = x0[4*q+0]*w00 + x3[4*q+0]*w01; o3.x = x0[4*q+0]*w30 + x3[4*q+0]*w31;
        o0.y = x0[4*q+1]*w00 + x3[4*q+1]*w01; o3.y = x0[4*q+1]*w30 + x3[4*q+1]*w31;
        o0.z = x0[4*q+2]*w00 + x3[4*q+2]*w01; o3.z = x0[4*q+2]*w30 + x3[4*q+2]*w31;
        o0.w = x0[4*q+3]*w00 + x3[4*q+3]*w01; o3.w = x0[4*q+3]*w30 + x3[4*q+3]*w31;
        ((float4*)p0)[q] = o0;
        ((float4*)p3)[q] = o3;
    }
}

// rel_out[r,o] = sum_i rel_emb[r,i]*prop_W[r,i,o] + prop_b[r,o]; 4x256 outputs
__global__ void rel_prop(const float* __restrict__ rel_emb, const float* __restrict__ prop_W,
                         const float* __restrict__ prop_b, float* __restrict__ outp)
{
    const int id = blockIdx.x * blockDim.x + threadIdx.x;
    const int r = id >> 8, o = id & 255;
    float s = 0.f;
#pragma unroll 8
    for (int i = 0; i < 64; ++i)
        s += rel_emb[r * 64 + i] * prop_W[(size_t)r * 64 * 256 + i * 256 + o];
    outp[id] = s + prop_b[id];
}

extern "C" void kernel_launch(void* const* d_in, const int* in_sizes, int n_in,
                              void* d_out, int out_size, void* d_ws, size_t ws_size,
                              hipStream_t stream)
{
    const float* feat      = (const float*)d_in[0];   // (4, N, 256)
    const float* rel_emb   = (const float*)d_in[1];   // (4, 64)
    const float* node_W    = (const float*)d_in[2];   // (3, 256, 256)
    const float* rel_W     = (const float*)d_in[3];   // (4, 64, 512)
    const float* res_W     = (const float*)d_in[4];   // (3, 256, 256)
    const float* res_b     = (const float*)d_in[5];   // (3, 256)
    const float* res_alpha = (const float*)d_in[6];   // (3,)
    const float* cross_at  = (const float*)d_in[7];   // (4, 8, 32)
    const float* prop_W    = (const float*)d_in[8];   // (4, 64, 256)
    const float* prop_b    = (const float*)d_in[9];   // (4, 256)
    const int*   src_idx   = (const int*)d_in[10];    // (4, E)
    const int*   dst_idx   = (const int*)d_in[11];    // (4, E)
    float* out = (float*)d_out;                       // node_out (4,N,256) ++ rel_out (4,256)

    // workspace carve-up (~115 MB), reused across the 4 relations
    float*    fsrc  = (float*)d_ws;
    float*    fdst  = fsrc  + (size_t)NNODES * HD;
    float*    eedge = fdst  + (size_t)NNODES * HD;
    float*    e_src = eedge + (size_t)NEDGES * HEADS;
    float*    e_dst = e_src + (size_t)NNODES * HEADS;
    float*    nsum  = e_dst + (size_t)NNODES * HEADS;
    unsigned* nmax  = (unsigned*)(nsum + (size_t)NNODES * HEADS);
    float*    ra    = (float*)(nmax + (size_t)NNODES * HEADS);

    const int NT_SRC[4] = {0, 1, 1, 2};
    const int NT_DST[4] = {1, 0, 2, 1};

    const dim3 ggrid(NNODES / 16);                // 3125 blocks x 128 threads (4 waves)
    const int NH = NNODES * HEADS;                // 400000
    const int EH = NEDGES * HEADS;                // 1600000
    const int bNH = (NH + 255) / 256;
    const int bEH = (EH + 255) / 256;
    const int bNC = (NNODES * HD + 255) / 256;

    for (int r = 0; r < 4; ++r) {
        const int s = NT_SRC[r], d = NT_DST[r];
        const float* fr = feat + (size_t)r * NNODES * INDIM;
        float* convr = out + (size_t)r * NNODES * HD;
        const int* srcr = src_idx + (size_t)r * NEDGES;
        const int* dstr = dst_idx + (size_t)r * NEDGES;

        gemm256_wmma<<<ggrid, 128, 0, stream>>>(fr, node_W + (size_t)s * 256 * 256, fsrc);
        gemm256_wmma<<<ggrid, 128, 0, stream>>>(fr, node_W + (size_t)d * 256 * 256, fdst);
        compute_ra<<<2, 256, 0, stream>>>(rel_emb + r * 64, rel_W + (size_t)r * 64 * 512, ra);
        compute_e<<<bNH, 256, 0, stream>>>(fsrc, fdst, ra, e_src, e_dst);
        fill_u32<<<bNH, 256, 0, stream>>>(nmax, 0u, NH);
        fill_f32<<<bNH, 256, 0, stream>>>(nsum, 0.f, NH);
        fill_f32<<<bNC, 256, 0, stream>>>(convr, 0.f, NNODES * HD);
        edge_logits<<<bEH, 256, 0, stream>>>(srcr, dstr, e_src, e_dst, eedge, nmax);
        edge_expsum<<<bEH, 256, 0, stream>>>(dstr, eedge, nmax, nsum);
        edge_scatter<<<bEH, 256, 0, stream>>>(srcr, dstr, fsrc, eedge, nsum, convr);
        combine_res_wmma<<<ggrid, 128, 0, stream>>>(fr, res_W + (size_t)d * 256 * 256,
                                                    res_b + d * 256, res_alpha, d, convr);
    }

    // cross-relation attention for dst ntype 1 (relations 0 and 3), in place
    cross_mix<<<bNH, 256, 0, stream>>>(out, out + (size_t)3 * NNODES * HD, cross_at);
    // relations 1 and 2 are singleton groups: conv_out already in place

    rel_prop<<<4, 256, 0, stream>>>(rel_emb, prop_W, prop_b, out + (size_t)4 * NNODES * HD);
}